// SoftmaxRefMatcher_11682311045217
// MI455X (gfx1250) — compile-verified
//
#include <hip/hip_runtime.h>
#include <hip/hip_bf16.h>

#define C 64
#define C2 32
#define N 256
#define WIDTH_ 256
#define HW (WIDTH_*WIDTH_)
#define NPAIR 6
// 1/TEMP * log2(e): weight = exp2(cos*KEXP - KEXP) == exp((cos-1)/TEMP)
#define KEXP 144.26950408889634f

typedef __attribute__((ext_vector_type(2))) float v2f;
typedef __attribute__((ext_vector_type(8))) float v8f;

// ---------------- zero the softmax-stat accumulators ----------------
__global__ void zero_acc_kernel(float* __restrict__ acc, int n) {
    int i = blockIdx.x * blockDim.x + threadIdx.x;
    if (i < n) acc[i] = 0.0f;
}

// ---- L2-normalize source frames (frames 0,4) into channel-PAIRED layout ----
// ws_src2[b][c/2][m] = (x_c, x_{c+1})  -> one b64 load per WMMA B operand
__global__ void prep_src_kernel(const float* __restrict__ dd, v2f* __restrict__ ws_src2) {
    int g = blockIdx.x * blockDim.x + threadIdx.x;   // 0 .. 2*HW-1
    int b = g >> 16;
    int m = g & (HW - 1);
    const float* srcf = dd + (size_t)(b * 4) * C * HW;
    float xs[C];
    float ss = 0.0f;
#pragma unroll
    for (int c = 0; c < C; ++c) { float x = srcf[(size_t)c * HW + m]; xs[c] = x; ss += x * x; }
    float inv = 1.0f / fmaxf(sqrtf(ss), 1e-12f);
    v2f* dst = ws_src2 + (size_t)b * C2 * HW;
#pragma unroll
    for (int c2 = 0; c2 < C2; ++c2) {
        v2f t;
        t.x = xs[2 * c2]     * inv;
        t.y = xs[2 * c2 + 1] * inv;
        dst[(size_t)c2 * HW + m] = t;
    }
}

// ---- L2-normalize target keypoint descriptors into channel-PAIRED layout ----
// ws_tgt2[p][c/2][n] = (x_c, x_{c+1})
__global__ void prep_tgt_kernel(const float* __restrict__ kd, v2f* __restrict__ ws_tgt2) {
    int g = blockIdx.x * blockDim.x + threadIdx.x;   // 0 .. 1535
    int pp = g >> 8;
    int n  = g & 255;
    int f  = (pp / 3) * 4 + 1 + (pp % 3);
    const float* src = kd + f * C * N;
    float xs[C];
    float ss = 0.0f;
#pragma unroll
    for (int c = 0; c < C; ++c) { float x = src[c * N + n]; xs[c] = x; ss += x * x; }
    float inv = 1.0f / fmaxf(sqrtf(ss), 1e-12f);
    v2f* dst = ws_tgt2 + pp * C2 * N;
#pragma unroll
    for (int c2 = 0; c2 < C2; ++c2) {
        v2f t;
        t.x = xs[2 * c2]     * inv;
        t.y = xs[2 * c2 + 1] * inv;
        dst[c2 * N + n] = t;
    }
}

// ---------------- fused WMMA cosine matmul + fixed-offset softmax statistics ----------------
// grid: 6 pairs * 8 kpt-tile-pairs (32 kpts each) * 32 pixel-chunks = 1536 blocks, 256 threads.
// Each B-register set (16 x b64) feeds TWO 16-wmma chains (32 kpts), halving L2 traffic.
__global__ void __launch_bounds__(256) match_wmma_kernel(const v2f* __restrict__ ws_src2,
                                                         const v2f* __restrict__ ws_tgt2,
                                                         float* __restrict__ ws_acc) {
    const int chunk = blockIdx.x & 31;
    const int ktp   = (blockIdx.x >> 5) & 7;   // pair of kpt tiles: kpts [ktp*32, ktp*32+32)
    const int p     = blockIdx.x >> 8;         // 0..5
    const int b     = p / 3;
    const int lane  = threadIdx.x & 31;
    const int wave  = threadIdx.x >> 5;
    const int lane15 = lane & 15;
    const int hi     = lane >> 4;              // lanes 0-15 -> K0/K1, lanes 16-31 -> K2/K3

    // Stage two 16x64 A-tiles in registers; reused for every pixel tile.
    const v2f* tg = ws_tgt2 + (size_t)p * C2 * N;
    v2f a0[16], a1[16];
#pragma unroll
    for (int kk = 0; kk < 16; ++kk) {
        int c2 = kk * 2 + hi;                  // channel pair (4kk+2hi, 4kk+2hi+1)
        a0[kk] = tg[c2 * N + ktp * 32 + lane15];
        a1[kk] = tg[c2 * N + ktp * 32 + 16 + lane15];
    }

    const v2f* sp = ws_src2 + (size_t)b * C2 * HW;

    float sw0[8], swu0[8], swv0[8], sw1[8], swu1[8], swv1[8];
#pragma unroll
    for (int r = 0; r < 8; ++r) {
        sw0[r] = swu0[r] = swv0[r] = 0.0f;
        sw1[r] = swu1[r] = swv1[r] = 0.0f;
    }

    // chunk = 2048 pixels = 128 tiles of 16; each wave owns 16 tiles
    for (int t = 0; t < 16; ++t) {
        int m0 = chunk * 2048 + (wave * 16 + t) * 16;
        int m  = m0 + lane15;
        float uf = (float)(m & (WIDTH_ - 1));
        float vf = (float)(m >> 8);

        v2f bm[16];
#pragma unroll
        for (int kk = 0; kk < 16; ++kk)
            bm[kk] = sp[(size_t)(kk * 2 + hi) * HW + m];   // global_load_b64

        v8f acc0 = {0.f, 0.f, 0.f, 0.f, 0.f, 0.f, 0.f, 0.f};
#pragma unroll
        for (int kk = 0; kk < 16; ++kk)
            acc0 = __builtin_amdgcn_wmma_f32_16x16x4_f32(
                false, a0[kk], false, bm[kk], (short)0, acc0, false, false);

        v8f acc1 = {0.f, 0.f, 0.f, 0.f, 0.f, 0.f, 0.f, 0.f};
#pragma unroll
        for (int kk = 0; kk < 16; ++kk)
            acc1 = __builtin_amdgcn_wmma_f32_16x16x4_f32(
                false, a1[kk], false, bm[kk], (short)0, acc1, false, false);

        // cos <= 1 exactly (unit vectors) -> fixed-offset softmax, no max pass.
        // single FMA + native base-2 exp per weight.
#pragma unroll
        for (int r = 0; r < 8; ++r) {
            float w0 = __builtin_amdgcn_exp2f(fmaf(acc0[r], KEXP, -KEXP));
            sw0[r]  += w0;
            swu0[r] += w0 * uf;
            swv0[r] += w0 * vf;
            float w1 = __builtin_amdgcn_exp2f(fmaf(acc1[r], KEXP, -KEXP));
            sw1[r]  += w1;
            swu1[r] += w1 * uf;
            swv1[r] += w1 * vf;
        }
    }

    // reduce across the 16 lanes sharing a kpt half (xor masks stay within each half)
#pragma unroll
    for (int r = 0; r < 8; ++r) {
#pragma unroll
        for (int mask = 1; mask <= 8; mask <<= 1) {
            sw0[r]  += __shfl_xor(sw0[r],  mask, 32);
            swu0[r] += __shfl_xor(swu0[r], mask, 32);
            swv0[r] += __shfl_xor(swv0[r], mask, 32);
            sw1[r]  += __shfl_xor(sw1[r],  mask, 32);
            swu1[r] += __shfl_xor(swu1[r], mask, 32);
            swv1[r] += __shfl_xor(swv1[r], mask, 32);
        }
    }
    if (lane15 == 0) {
        int kbase = hi * 8;
#pragma unroll
        for (int r = 0; r < 8; ++r) {
            int row0 = p * N + ktp * 32 + kbase + r;
            unsafeAtomicAdd(&ws_acc[row0 * 3 + 0], sw0[r]);
            unsafeAtomicAdd(&ws_acc[row0 * 3 + 1], swu0[r]);
            unsafeAtomicAdd(&ws_acc[row0 * 3 + 2], swv0[r]);
            int row1 = row0 + 16;
            unsafeAtomicAdd(&ws_acc[row1 * 3 + 0], sw1[r]);
            unsafeAtomicAdd(&ws_acc[row1 * 3 + 1], swu1[r]);
            unsafeAtomicAdd(&ws_acc[row1 * 3 + 2], swv1[r]);
        }
    }
}

// ---------------- finalize: soft-argmax coords, score gather, id arrays ----------------
__global__ void finalize_kernel(const float* __restrict__ ws_acc,
                                const float* __restrict__ scores,
                                float* __restrict__ out) {
    int row = blockIdx.x * blockDim.x + threadIdx.x;   // 0..1535
    int pp = row >> 8;
    int n  = row & 255;
    float s   = ws_acc[row * 3 + 0];
    float su  = ws_acc[row * 3 + 1];
    float sv  = ws_acc[row * 3 + 2];
    float inv = 1.0f / fmaxf(s, 1e-30f);
    out[row * 2 + 0] = su * inv;                       // pseudo u
    out[row * 2 + 1] = sv * inv;                       // pseudo v
    int tgt = (pp / 3) * 4 + 1 + (pp % 3);
    out[NPAIR * N * 2 + pp * N + n] = scores[tgt * N + n];
    if (row < NPAIR) {
        int t2 = (row / 3) * 4 + 1 + (row % 3);
        out[NPAIR * N * 3 + row]         = (float)t2;              // tgt_ids
        out[NPAIR * N * 3 + NPAIR + row] = (float)((row / 3) * 4); // src_ids
    }
}

extern "C" void kernel_launch(void* const* d_in, const int* in_sizes, int n_in,
                              void* d_out, int out_size, void* d_ws, size_t ws_size,
                              hipStream_t stream) {
    const float* scores = (const float*)d_in[0];   // (8,1,256)
    const float* kd     = (const float*)d_in[1];   // (8,64,256)
    const float* dd     = (const float*)d_in[2];   // (8,64,256,256)
    // d_in[3] (keypoint_coords) unused by the reference outputs

    float* ws      = (float*)d_ws;
    v2f*   ws_src2 = (v2f*)ws;                                   // 2*32*65536 pairs (33.5 MB)
    v2f*   ws_tgt2 = ws_src2 + (size_t)2 * C2 * HW;              // 6*32*256 pairs
    float* ws_acc  = (float*)(ws_tgt2 + (size_t)NPAIR * C2 * N); // 1536*3 floats
    float* out     = (float*)d_out;

    zero_acc_kernel<<<dim3(18), dim3(256), 0, stream>>>(ws_acc, NPAIR * N * 3);
    prep_src_kernel<<<dim3(512), dim3(256), 0, stream>>>(dd, ws_src2);
    prep_tgt_kernel<<<dim3(6), dim3(256), 0, stream>>>(kd, ws_tgt2);
    match_wmma_kernel<<<dim3(1536), dim3(256), 0, stream>>>(ws_src2, ws_tgt2, ws_acc);
    finalize_kernel<<<dim3(6), dim3(256), 0, stream>>>(ws_acc, scores, out);
}